// BiLSTMCRF_58798102282395
// MI455X (gfx1250) — compile-verified
//
#include <hip/hip_runtime.h>

// Problem constants from the reference (setup_inputs): char_feats [L,B,D] f32,
// word_ids [B,L] i32 (monotone, -1 at CLS/SEP), attention_mask [B,L] i32.
// Output: word_feats [MW,B,D] f32  ++  masks [MW,B] (as f32 0/1).
constexpr int B  = 64;
constexpr int L  = 512;
constexpr int D  = 768;
constexpr int MW = 256;
constexpr int MT = MW / 16;   // 16 word-tiles of 16 words
constexpr int NT = D / 16;    // 48 feature-tiles of 16 dims
constexpr int WAVES_PER_BLOCK = 8;   // 256 threads, wave32

typedef float v2f __attribute__((ext_vector_type(2)));
typedef float v8f __attribute__((ext_vector_type(8)));

// d_ws int layout: [0,B): word_num ; [B,2B): char_num ; [2B .. 2B+B*MT*2): {lo,hi} per (b,tile)
__global__ void wordpool_prep(const int* __restrict__ wid,
                              const int* __restrict__ amask,
                              int* __restrict__ ws) {
  __shared__ int s_lo[MT], s_hi[MT];
  __shared__ int s_wmax, s_cnt;
  const int b   = blockIdx.x;
  const int tid = threadIdx.x;
  if (tid < MT) { s_lo[tid] = 0x7fffffff; s_hi[tid] = 0; }
  if (tid == 0) { s_wmax = -1; s_cnt = 0; }
  __syncthreads();
  int csum = 0;
  for (int pos = tid; pos < L; pos += blockDim.x) {
    const int w = wid[b * L + pos];
    csum += amask[b * L + pos];
    if (w >= 0) {
      atomicMin(&s_lo[w >> 4], pos);      // ds atomics: char range per 16-word tile
      atomicMax(&s_hi[w >> 4], pos + 1);
      atomicMax(&s_wmax, w);
    }
  }
  atomicAdd(&s_cnt, csum);
  __syncthreads();
  if (tid == 0) { ws[b] = s_wmax + 1; ws[B + b] = s_cnt - 2; }
  if (tid < MT) {
    ws[2 * B + (b * MT + tid) * 2 + 0] = s_lo[tid];
    ws[2 * B + (b * MT + tid) * 2 + 1] = s_hi[tid];
  }
}

// One wave computes a 16(words) x 16(dims) output tile for one batch via
// exact-fp32 one-hot GEMM on V_WMMA_F32_16X16X4_F32. A second WMMA against an
// all-ones B matrix accumulates the per-word counts directly in C layout.
__global__ void wordpool_wmma(const float* __restrict__ feats,   // [L,B,D]
                              const int*   __restrict__ wid,     // [B,L]
                              const int*   __restrict__ ws,
                              float* __restrict__ out) {
  const int lane = threadIdx.x & 31;
  const int wave = blockIdx.x * (blockDim.x >> 5) + (threadIdx.x >> 5);
  const int b  = wave / (MT * NT);
  const int r  = wave % (MT * NT);
  const int t  = r / NT;        // word tile
  const int nt = r % NT;        // feature tile

  // wave-uniform scalars (force SGPRs so the k-loop is uniform -> EXEC all 1s)
  const int lo   = __builtin_amdgcn_readfirstlane(ws[2 * B + (b * MT + t) * 2 + 0]);
  const int hi   = __builtin_amdgcn_readfirstlane(ws[2 * B + (b * MT + t) * 2 + 1]);
  const int wnum = __builtin_amdgcn_readfirstlane(ws[b]);
  const int cnum = __builtin_amdgcn_readfirstlane(ws[B + b]);

  const int m   = lane & 15;          // A-matrix row (word within tile)
  const int n   = lane & 15;          // B/C-matrix column (dim within tile)
  const int kk0 = (lane >> 4) * 2;    // K sub-index pair held by this lane half
  const int m0  = t * 16;
  const int n0  = nt * 16;

  v8f csum = {};                      // 16x16 f32 sums accumulator
  v8f ccnt = {};                      // 16x16 f32 counts accumulator
  v2f bones; bones.x = 1.0f; bones.y = 1.0f;   // all-ones B matrix

  const int* __restrict__ wrow = wid + b * L;
  for (int k = (lo & ~3); k < hi; k += 4) {
    const int p0 = k + kk0, p1 = p0 + 1;
    const int q0 = min(p0, L - 1), q1 = min(p1, L - 1);
    // one-hot A per ISA layout: lanes 0-15 -> K={0,1}, lanes 16-31 -> K={2,3}
    const int w0 = wrow[q0], w1 = wrow[q1];
    v2f a;
    a.x = (w0 == m0 + m && p0 >= 1 && p0 <= cnum) ? 1.0f : 0.0f;
    a.y = (w1 == m0 + m && p1 >= 1 && p1 <= cnum) ? 1.0f : 0.0f;
    // B = feats rows k..k+3, 16 dims; lane half selects K pair, lane&15 = N
    v2f bm;
    bm.x = feats[(q0 * B + b) * D + n0 + n];
    bm.y = feats[(q1 * B + b) * D + n0 + n];
    // CDNA5 prefetch of the next K-step's rows (global_prefetch_b8)
    __builtin_prefetch(&feats[(min(p0 + 4, L - 1) * B + b) * D + n0 + n], 0, 1);
    csum = __builtin_amdgcn_wmma_f32_16x16x4_f32(false, a, false, bm,
                                                 (short)0, csum, false, false);
    ccnt = __builtin_amdgcn_wmma_f32_16x16x4_f32(false, a, false, bones,
                                                 (short)0, ccnt, false, false);
  }

  // C layout: VGPR i -> M = i (lanes 0-15) / i+8 (lanes 16-31), N = lane&15.
  // Counts are small positive integers: one v_rcp_f32 + v_mul replaces the
  // full IEEE divide expansion; w<wnum folds to v_cndmask (no divergence).
  const int hihalf = lane >> 4;
#pragma unroll
  for (int i = 0; i < 8; ++i) {
    const int w     = m0 + i + 8 * hihalf;
    const float c   = fmaxf(ccnt[i], 1.0f);
    const float inv = __builtin_amdgcn_rcpf(c);
    const float v   = (w < wnum) ? csum[i] * inv : 0.0f;
    out[(w * B + b) * D + n0 + n] = v;   // coalesced: 16 lanes -> 64B rows
  }
}

__global__ void wordpool_mask(const int* __restrict__ ws, float* __restrict__ out) {
  const int idx = blockIdx.x * blockDim.x + threadIdx.x;   // over MW*B
  if (idx >= MW * B) return;
  const int w = idx / B, b = idx % B;
  out[MW * B * D + idx] = (w < ws[b]) ? 1.0f : 0.0f;       // masks[w][b]
}

extern "C" void kernel_launch(void* const* d_in, const int* in_sizes, int n_in,
                              void* d_out, int out_size, void* d_ws, size_t ws_size,
                              hipStream_t stream) {
  (void)in_sizes; (void)n_in; (void)out_size; (void)ws_size;
  const float* feats = (const float*)d_in[0];   // char_feats [L,B,D] f32
  const int*   wid   = (const int*)d_in[1];     // word_ids   [B,L]  i32
  const int*   amask = (const int*)d_in[2];     // attention_mask [B,L] i32
  float* out = (float*)d_out;
  int*   ws  = (int*)d_ws;

  wordpool_prep<<<B, 256, 0, stream>>>(wid, amask, ws);

  const int waves  = B * MT * NT;               // 49152 tiles
  const int blocks = waves / WAVES_PER_BLOCK;   // 6144 blocks of 8 waves
  wordpool_wmma<<<blocks, WAVES_PER_BLOCK * 32, 0, stream>>>(feats, wid, ws, out);

  wordpool_mask<<<(MW * B + 255) / 256, 256, 0, stream>>>(ws, out);
}